// BinaryTokenClassificationModel_58377195487334
// MI455X (gfx1250) — compile-verified
//
#include <hip/hip_runtime.h>

typedef __attribute__((ext_vector_type(2))) float v2f;
typedef __attribute__((ext_vector_type(4))) float v4f;
typedef __attribute__((ext_vector_type(8))) float v8f;

// Problem shape (fixed by the harness setup)
#define BB 16
#define LL 2048
#define HH 1024
#define WW 1024
#define BL (BB * LL)   // 32768 tokens
#define BW (BB * WW)   // 16384 (batch, word) slots

// Workspace layout (floats):
//   [0*BW, 1*BW) src sum  -> src_proj after finalize
//   [1*BW, 2*BW) tgt sum  -> tgt_proj after finalize
//   [2*BW, 3*BW) src count
//   [3*BW, 4*BW) tgt count

__global__ __launch_bounds__(256) void zero_ws_kernel(float* __restrict__ ws) {
    int i = blockIdx.x * 256 + threadIdx.x;
    if (i < 4 * BW) ws[i] = 0.0f;
}

// One wave = 16 tokens. D = A(16x4 fp32 of hidden) x B(4x16, cols 0/1 = w_s/w_t)
// accumulated over K=1024 with v_wmma_f32_16x16x4_f32.
// B panel staged in LDS with a zero-padded region so ALL lanes issue one
// unconditional ds_load_b64 per step (no exec-masked loads in the hot loop).
// D column 0 = ps (lanes 0/16), column 1 = pt (lanes 1/17). Lanes 2,3,18,19
// contribute the segment-count histogram (their D columns are exactly zero).
__global__ __launch_bounds__(256)
void proj_pool_kernel(const float* __restrict__ hidden,
                      const float* __restrict__ cw,
                      const int*   __restrict__ sids,
                      const int*   __restrict__ tids,
                      float*       __restrict__ ws) {
    __shared__ float s_bpad[4 * HH];               // 16 KB: w_s | w_t | 0s | 0s
    for (int i = threadIdx.x; i < 2 * HH; i += 256) s_bpad[i] = cw[i];
    for (int i = threadIdx.x; i < 2 * HH; i += 256) s_bpad[2 * HH + i] = 0.0f;
    __syncthreads();

    const int lane   = threadIdx.x & 31;           // wave32
    const int wave   = (blockIdx.x * 256 + threadIdx.x) >> 5;
    const int token0 = wave << 4;                  // 16 tokens per wave
    const int n      = lane & 15;                  // A: row M, B: col N
    const int kh     = (lane >> 4) << 1;           // K sub-offset: 0 or 2

    const float* arow = hidden + (size_t)(token0 + n) * HH + kh;
    // lanes n>=2 read the zeroed LDS region -> zero B columns, no branches
    const float* brow = s_bpad + (n < 2 ? n : 2) * HH + kh;

    v8f acc = {};
#pragma unroll 8
    for (int k = 0; k < HH; k += 4) {
        v2f a = *(const v2f*)(arow + k);           // global_load_b64
        v2f b = *(const v2f*)(brow + k);           // ds_load_b64, unconditional
        acc = __builtin_amdgcn_wmma_f32_16x16x4_f32(
            /*neg_a=*/false, a, /*neg_b=*/false, b,
            /*c_mod=*/(short)0, acc, /*reuse_a=*/false, /*reuse_b=*/false);
    }

    // C/D layout: lane<16 has col=lane, rows 0..7 in acc[0..7];
    //             lane>=16 has col=lane-16, rows 8..15.
    if ((lane & 12) == 0) {                        // lanes 0-3 and 16-19
        const int  col   = lane & 3;               // 0:psum 1:tsum 2:pcnt 3:tcnt
        const int  rbase = (lane >> 4) << 3;
        const int* ids   = (col & 1) ? tids : sids;
        float* sums = ws + ((col & 1) ? BW : 0);
        float* cnts = ws + 2 * BW + ((col & 1) ? BW : 0);
#pragma unroll
        for (int r = 0; r < 8; ++r) {
            int tok = token0 + rbase + r;
            int b_  = tok / LL;
            int id  = ids[tok];
            if (id >= 0) {
                if (col < 2) atomicAdd(&sums[b_ * WW + id], acc[r]);
                else         atomicAdd(&cnts[b_ * WW + id], 1.0f);
            }
        }
    }
}

__global__ __launch_bounds__(256) void finalize_kernel(float* __restrict__ ws) {
    int i = blockIdx.x * 256 + threadIdx.x;
    if (i < BW) {
        ws[i]      = ws[i]      / fmaxf(ws[2 * BW + i], 1.0f);
        ws[BW + i] = ws[BW + i] / fmaxf(ws[3 * BW + i], 1.0f);
    }
}

// logits[b,s,t] = src_proj[b,s] + tgt_proj[b,t] + bias. 64 MiB streamed out
// once: 4 floats/thread, non-temporal b128 stores.
__global__ __launch_bounds__(256)
void outer_sum_kernel(const float* __restrict__ ws,
                      const float* __restrict__ bias,
                      float*       __restrict__ out) {
    int idx = blockIdx.x * 256 + threadIdx.x;      // over B*W*W/4
    int t4  = idx & (WW / 4 - 1);
    int s   = (idx >> 8) & (WW - 1);
    int b   = idx >> 18;                           // W*W/4 = 2^18 per batch
    float sp = ws[b * WW + s] + bias[0];
    v4f tv = *(const v4f*)(ws + BW + b * WW + t4 * 4);
    v4f o  = tv + sp;
    __builtin_nontemporal_store(o, (v4f*)out + idx);
}

extern "C" void kernel_launch(void* const* d_in, const int* in_sizes, int n_in,
                              void* d_out, int out_size, void* d_ws, size_t ws_size,
                              hipStream_t stream) {
    const float* hidden = (const float*)d_in[0];
    const float* cw     = (const float*)d_in[1];
    const float* cb     = (const float*)d_in[2];
    const int*   sids   = (const int*)d_in[3];
    const int*   tids   = (const int*)d_in[4];
    float* ws  = (float*)d_ws;
    float* out = (float*)d_out;

    zero_ws_kernel<<<(4 * BW + 255) / 256, 256, 0, stream>>>(ws);
    // 2048 waves * 32 lanes / 256 threads = 256 blocks
    proj_pool_kernel<<<(BL / 16 * 32) / 256, 256, 0, stream>>>(hidden, cw, sids, tids, ws);
    finalize_kernel<<<(BW + 255) / 256, 256, 0, stream>>>(ws);
    outer_sum_kernel<<<(BB * WW * WW / 4) / 256, 256, 0, stream>>>(ws, cb, out);
}